// Cif_59914793779623
// MI455X (gfx1250) — compile-verified
//
#include <hip/hip_runtime.h>
#include <math.h>

// CIF (Continuous Integrate-and-Fire) for MI455X / gfx1250.
// T=2048, B=32, E=512, DENSE=512.
//
// Kernel 1: fused f32 WMMA GEMM (16x16x4_f32, exact precision for the
//           threshold test) -> relu -> dot(wproj) -> sigmoid -> weight[b,t]
// Kernel 2: 32-thread scalar integrate-and-fire scan over weights producing,
//           per compacted output slot: [lo,hi] time range, opening coef,
//           scale (1.0 or tail 1/(w+eps)); plus fired_marks and quantity.
// Kernel 3: one block per output frame: cif_out[b,s,:] = scale *
//           sum_{t=lo..hi} coef_t * enc[t,b,:], written directly compacted.

#define T_ 2048
#define B_ 32
#define E_ 512
#define D_ 512
#define THRESH_ 1.0f
#define TAIL_MIN_W_ 0.6f
#define EPS_ 1e-10f

typedef __attribute__((ext_vector_type(2))) float v2f;
typedef __attribute__((ext_vector_type(8))) float v8f;

// Only meaningful on the device pass; host pass may not report aux builtins.
#if defined(__HIP_DEVICE_COMPILE__) && !__has_builtin(__builtin_amdgcn_wmma_f32_16x16x4_f32)
#error "gfx1250 wmma f32 16x16x4 builtin not available on this toolchain (device pass)"
#endif

// ---------------------------------------------------------------------------
// Kernel 1: weights via f32 WMMA.
// Grid: 512 blocks x 256 threads (8 waves). One wave handles one M-tile of
// 16 consecutive t-rows of one batch. M = b*T + t (T divisible by 16, so a
// tile never crosses batches).
// ---------------------------------------------------------------------------
__global__ __launch_bounds__(256) void cif_weight_gemm(
    const float* __restrict__ enc,          // [T,B,E]  (x[b,t,e] = enc[t,b,e])
    const unsigned char* __restrict__ pmask,// [B,T] bool
    const float* __restrict__ Wd,           // [E,D]
    const float* __restrict__ bd,           // [D]
    const float* __restrict__ wp,           // [D,1]
    const float* __restrict__ wpb,          // [1]
    float* __restrict__ weight)             // [B,T]
{
    const int wave = threadIdx.x >> 5;
    const int lane = threadIdx.x & 31;
    const int half = lane >> 4;      // 0: lanes 0-15, 1: lanes 16-31
    const int l16  = lane & 15;
    const int mtile = blockIdx.x * 8 + wave;   // 0..4095
    const int m0 = mtile << 4;
    const int b  = m0 / T_;
    const int t0 = m0 % T_;

    // A-fragment row for this lane (ISA 7.12.2: lanes 0-15 rows M=0-15 K=0,1;
    // lanes 16-31 same rows, K=2,3)
    const float* Arow = enc + ((size_t)(t0 + l16) * B_ + b) * E_;
    const int kbase = half * 2;

    v8f rowacc = {0.f,0.f,0.f,0.f,0.f,0.f,0.f,0.f};

    for (int n0 = 0; n0 < D_; n0 += 16) {
        v8f c = {0.f,0.f,0.f,0.f,0.f,0.f,0.f,0.f};
        const float* Wcol = Wd + n0 + l16;   // column n = n0 + l16
        #pragma unroll 4
        for (int k0 = 0; k0 < E_; k0 += 4) {
            const int kk = k0 + kbase;
            v2f a, bb;
            a.x  = Arow[kk];
            a.y  = Arow[kk + 1];
            bb.x = Wcol[(size_t)kk * D_];        // B[k][n], K rows split like A
            bb.y = Wcol[(size_t)(kk + 1) * D_];
            c = __builtin_amdgcn_wmma_f32_16x16x4_f32(
                    false, a, false, bb, (short)0, c, false, false);
        }
        // C layout: VGPR r = row (r + 8*half), column n0+l16.
        const float bias = bd[n0 + l16];
        const float wpn  = wp[n0 + l16];
        #pragma unroll
        for (int r = 0; r < 8; ++r) {
            float h = c[r] + bias;
            h = h > 0.f ? h : 0.f;               // relu
            rowacc[r] += h * wpn;                // partial dot with wproj
        }
    }

    // Reduce over the 16 lanes holding different columns (xor within half-wave)
    #pragma unroll
    for (int r = 0; r < 8; ++r) {
        float v = rowacc[r];
        v += __shfl_xor(v, 1, 32);
        v += __shfl_xor(v, 2, 32);
        v += __shfl_xor(v, 4, 32);
        v += __shfl_xor(v, 8, 32);
        rowacc[r] = v;
    }

    if (l16 == 0) {  // lane 0 -> rows 0-7, lane 16 -> rows 8-15
        const float wb = wpb[0];
        #pragma unroll
        for (int r = 0; r < 8; ++r) {
            const int tt = t0 + half * 8 + r;
            const float z = rowacc[r] + wb;
            float wgt = 1.0f / (1.0f + expf(-z));     // sigmoid
            if (pmask[(size_t)b * T_ + tt]) wgt = 0.f; // mask padding
            weight[(size_t)b * T_ + tt] = wgt;
        }
    }
}

// ---------------------------------------------------------------------------
// Kernel 2: scalar integrate-and-fire scan, one thread per batch.
// Emits per output slot s: segLo, segHi, coefLo (opening coefficient),
// segScale (1.0 for fired frames, 1/(w+eps) for the tail frame), and per
// (b,t): coefSelf (interior/closing coefficient), fired_marks, quantity.
// padding_mask is monotonic, so the first padded t == length[b] (tail step).
// ---------------------------------------------------------------------------
__global__ void cif_scan(
    const float* __restrict__ weight,        // [B,T]
    const unsigned char* __restrict__ pmask, // [B,T]
    float* __restrict__ coefSelf,            // [B,T]
    int*   __restrict__ segLo,               // [B,T]
    int*   __restrict__ segHi,               // [B,T]
    float* __restrict__ coefLo,              // [B,T]
    float* __restrict__ segScale,            // [B,T]
    int*   __restrict__ segCount,            // [B]
    float* __restrict__ marks_out,           // [B,T] (float 0/1)
    float* __restrict__ qty_out)             // [B]
{
    const int b = blockIdx.x * blockDim.x + threadIdx.x;
    if (b >= B_) return;
    const float* wrow = weight + (size_t)b * T_;
    const unsigned char* mrow = pmask + (size_t)b * T_;
    float* crow = coefSelf + (size_t)b * T_;
    float* markrow = marks_out + (size_t)b * T_;
    const size_t sb = (size_t)b * T_;

    float prev_w = 0.f, qty = 0.f;
    int seg = 0;
    int cur_lo = 0;
    float cur_cl = 0.f;
    bool tailDone = false;

    for (int t = 0; t < T_; ++t) {
        float mark = 0.f;
        const float w = wrow[t];
        if (mrow[t]) {
            // first padded step == padding_start: tail frame
            if (!tailDone) {
                tailDone = true;
                if (prev_w > TAIL_MIN_W_) {   // cur_w = prev_w (w == 0 here)
                    segLo[sb + seg] = cur_lo;
                    segHi[sb + seg] = t - 1;
                    coefLo[sb + seg] = cur_cl;
                    segScale[sb + seg] = 1.0f / (prev_w + EPS_);
                    mark = 1.f;
                    ++seg;
                }
            }
            crow[t] = 0.f;
        } else {
            qty += w;
            const bool fired = (prev_w + w) >= THRESH_;
            if (fired) {
                const float remained = 1.0f - prev_w;
                crow[t] = remained;               // closing coefficient
                segLo[sb + seg] = cur_lo;
                segHi[sb + seg] = t;
                coefLo[sb + seg] = cur_cl;
                segScale[sb + seg] = 1.0f;
                mark = 1.f;
                ++seg;
                cur_lo = t;                        // new segment opens at t
                cur_cl = w - remained;             // carry coefficient
                prev_w = w - remained;
            } else {
                crow[t] = w;
                if (t == cur_lo) cur_cl = w;       // segment 0 start
                prev_w += w;
            }
        }
        markrow[t] = mark;
    }
    segCount[b] = seg;
    qty_out[b] = qty;
}

// ---------------------------------------------------------------------------
// Kernel 3: build compacted cif_out. One block (256 threads) per (b, slot);
// each thread owns 2 of the 512 channels. Slots >= count are zero-filled
// (d_out is poisoned by the harness).
// ---------------------------------------------------------------------------
__global__ __launch_bounds__(256) void cif_gather(
    const float* __restrict__ enc,       // [T,B,E]
    const float* __restrict__ coefSelf,  // [B,T]
    const int*   __restrict__ segLo,
    const int*   __restrict__ segHi,
    const float* __restrict__ coefLo,
    const float* __restrict__ segScale,
    const int*   __restrict__ segCount,
    float* __restrict__ cif_out,         // [B,T,E]
    float* __restrict__ mask_out)        // [B,T] (float 0/1)
{
    const int blk = blockIdx.x;          // = b*T + s
    const int b = blk / T_;
    const int s = blk % T_;
    const int tid = threadIdx.x;
    float* outfr = cif_out + (size_t)blk * E_;

    if (s >= segCount[b]) {
        outfr[tid] = 0.f;
        outfr[tid + 256] = 0.f;
        if (tid == 0) mask_out[blk] = 0.f;
        return;
    }
    const size_t sb = (size_t)b * T_ + s;
    const int lo = segLo[sb];
    const int hi = segHi[sb];
    const float cl = coefLo[sb];
    const float sc = segScale[sb];

    float acc0 = 0.f, acc1 = 0.f;
    for (int t = lo; t <= hi; ++t) {
        const float coef = (t == lo) ? cl : coefSelf[(size_t)b * T_ + t];
        const float* xr = enc + ((size_t)t * B_ + b) * E_;
        acc0 += coef * xr[tid];
        acc1 += coef * xr[tid + 256];
    }
    outfr[tid]       = acc0 * sc;
    outfr[tid + 256] = acc1 * sc;
    if (tid == 0) mask_out[blk] = 1.f;
}

// ---------------------------------------------------------------------------
extern "C" void kernel_launch(void* const* d_in, const int* in_sizes, int n_in,
                              void* d_out, int out_size, void* d_ws, size_t ws_size,
                              hipStream_t stream) {
    const float*         enc   = (const float*)d_in[0];         // [T,B,E]
    const unsigned char* pmask = (const unsigned char*)d_in[1]; // [B,T] bool
    const float*         Wd    = (const float*)d_in[2];         // [E,D]
    const float*         bd    = (const float*)d_in[3];         // [D]
    const float*         wp    = (const float*)d_in[4];         // [D,1]
    const float*         wpb   = (const float*)d_in[5];         // [1]

    // Output tuple, concatenated flat in return order (all as f32):
    // cif_out [B,T,E] | cif_out_padding_mask [B,T] | quantity_out [B] | fired_marks [B,T]
    float* out     = (float*)d_out;
    float* cif     = out;
    float* maskpad = out + (size_t)B_ * T_ * E_;
    float* qty     = maskpad + (size_t)B_ * T_;
    float* marks   = qty + B_;

    // Workspace carve-up (~1.5 MB)
    char* ws = (char*)d_ws;
    float* weight   = (float*)ws; ws += (size_t)B_ * T_ * sizeof(float);
    float* coefSelf = (float*)ws; ws += (size_t)B_ * T_ * sizeof(float);
    int*   segLo    = (int*)ws;   ws += (size_t)B_ * T_ * sizeof(int);
    int*   segHi    = (int*)ws;   ws += (size_t)B_ * T_ * sizeof(int);
    float* coefLo   = (float*)ws; ws += (size_t)B_ * T_ * sizeof(float);
    float* segScale = (float*)ws; ws += (size_t)B_ * T_ * sizeof(float);
    int*   segCount = (int*)ws;   ws += B_ * sizeof(int);

    // 1) weights via f32 WMMA (4096 M-tiles, 8 waves/block)
    cif_weight_gemm<<<512, 256, 0, stream>>>(enc, pmask, Wd, bd, wp, wpb, weight);
    // 2) scalar integrate-and-fire scan (one thread per batch)
    cif_scan<<<1, 32, 0, stream>>>(weight, pmask, coefSelf, segLo, segHi,
                                   coefLo, segScale, segCount, marks, qty);
    // 3) compacted output frames
    cif_gather<<<B_ * T_, 256, 0, stream>>>(enc, coefSelf, segLo, segHi,
                                            coefLo, segScale, segCount,
                                            cif, maskpad);
}